// GraphSAGEEncoder_33432025432488
// MI455X (gfx1250) — compile-verified
//
#include <hip/hip_runtime.h>
#include <hip/hip_bf16.h>

typedef __attribute__((ext_vector_type(2))) float v2f;
typedef __attribute__((ext_vector_type(8))) float v8f;

// ---------------------------------------------------------------------------
// Small utility kernels
// ---------------------------------------------------------------------------
__global__ void zero_int_kernel(int* __restrict__ p, int n) {
    int i = blockIdx.x * blockDim.x + threadIdx.x;
    if (i < n) p[i] = 0;
}

__global__ void hist_kernel(const long long* __restrict__ dst,
                            int* __restrict__ cnt, int E) {
    int e = blockIdx.x * blockDim.x + threadIdx.x;
    if (e < E) atomicAdd(&cnt[(int)dst[e]], 1);
}

// Per-1024-chunk exclusive scan; emits chunk totals.
__global__ void __launch_bounds__(1024)
scan_block_kernel(const int* __restrict__ cnt, int* __restrict__ offs,
                  int* __restrict__ blocksum, int n) {
    __shared__ int tmp[1024];
    int tid = threadIdx.x;
    int gid = blockIdx.x * 1024 + tid;
    int v = (gid < n) ? cnt[gid] : 0;
    tmp[tid] = v;
    __syncthreads();
    for (int off = 1; off < 1024; off <<= 1) {
        int t = (tid >= off) ? tmp[tid - off] : 0;
        __syncthreads();
        tmp[tid] += t;
        __syncthreads();
    }
    if (gid < n) offs[gid] = tmp[tid] - v;          // exclusive
    if (tid == 1023) blocksum[blockIdx.x] = tmp[tid];
}

// Single-block exclusive scan of chunk totals (nb <= 128).
__global__ void __launch_bounds__(128)
scan_top_kernel(int* __restrict__ blocksum, int nb) {
    __shared__ int tmp[128];
    int tid = threadIdx.x;
    int v = (tid < nb) ? blocksum[tid] : 0;
    tmp[tid] = v;
    __syncthreads();
    for (int off = 1; off < 128; off <<= 1) {
        int t = (tid >= off) ? tmp[tid - off] : 0;
        __syncthreads();
        tmp[tid] += t;
        __syncthreads();
    }
    if (tid < nb) blocksum[tid] = tmp[tid] - v;     // exclusive
}

__global__ void __launch_bounds__(1024)
scan_add_kernel(int* __restrict__ offs, const int* __restrict__ blocksum,
                int n) {
    int gid = blockIdx.x * 1024 + threadIdx.x;
    if (gid < n) offs[gid] += blocksum[blockIdx.x];
}

// Bucket edges by destination: eidx holds source node ids, CSR order.
__global__ void place_kernel(const long long* __restrict__ src,
                             const long long* __restrict__ dst,
                             const int* __restrict__ offs,
                             int* __restrict__ cursor,
                             int* __restrict__ eidx, int E) {
    int e = blockIdx.x * blockDim.x + threadIdx.x;
    if (e < E) {
        int d = (int)dst[e];
        int p = atomicAdd(&cursor[d], 1);
        eidx[offs[d] + p] = (int)src[e];
    }
}

// Interleave K-pairs so a B fragment is one aligned 8-byte load:
// P[(k/2)*NC*2 + c*2 + (k&1)] = W[k*NC + c]
__global__ void repack_kernel(const float* __restrict__ W,
                              float* __restrict__ P, int total, int NC) {
    int i = blockIdx.x * blockDim.x + threadIdx.x;
    if (i < total) {
        int k = i / NC, c = i % NC;
        P[(size_t)(k >> 1) * NC * 2 + c * 2 + (k & 1)] = W[i];
    }
}

// ---------------------------------------------------------------------------
// CSR mean-aggregation: one wave per node. Lane gathers float4 chunks of each
// neighbor row (coalesced b128 loads, L2-resident), accumulates in registers,
// writes the mean with plain b128 stores. No float atomics at all.
// ---------------------------------------------------------------------------
template <int F>
__global__ void __launch_bounds__(256)
aggregate_kernel(const float* __restrict__ x, const int* __restrict__ eidx,
                 const int* __restrict__ offs, const int* __restrict__ cnt,
                 float* __restrict__ agg, int N) {
    int node = blockIdx.x * (blockDim.x >> 5) + (threadIdx.x >> 5);
    if (node >= N) return;
    int lane = threadIdx.x & 31;
    int deg = cnt[node];
    int start = offs[node];

    constexpr int C = F / 128;  // float4 chunks per lane
    float4 acc[C];
#pragma unroll
    for (int c = 0; c < C; ++c) acc[c] = make_float4(0.f, 0.f, 0.f, 0.f);

    int j = 0;
    for (; j + 1 < deg; j += 2) {  // 2-way unroll to overlap gathers
        int s0 = eidx[start + j];
        int s1 = eidx[start + j + 1];
        const float4* x0 = (const float4*)(x + (size_t)s0 * F);
        const float4* x1 = (const float4*)(x + (size_t)s1 * F);
#pragma unroll
        for (int c = 0; c < C; ++c) {
            float4 v0 = x0[lane + 32 * c];
            float4 v1 = x1[lane + 32 * c];
            acc[c].x += v0.x + v1.x;
            acc[c].y += v0.y + v1.y;
            acc[c].z += v0.z + v1.z;
            acc[c].w += v0.w + v1.w;
        }
    }
    if (j < deg) {
        int s0 = eidx[start + j];
        const float4* x0 = (const float4*)(x + (size_t)s0 * F);
#pragma unroll
        for (int c = 0; c < C; ++c) {
            float4 v0 = x0[lane + 32 * c];
            acc[c].x += v0.x;
            acc[c].y += v0.y;
            acc[c].z += v0.z;
            acc[c].w += v0.w;
        }
    }

    float scale = (deg > 0) ? (1.0f / (float)deg) : 1.0f;  // max(cnt,1)
    float4* ag = (float4*)(agg + (size_t)node * F);
#pragma unroll
    for (int c = 0; c < C; ++c) {
        float4 r;
        r.x = acc[c].x * scale;
        r.y = acc[c].y * scale;
        r.z = acc[c].z * scale;
        r.w = acc[c].w * scale;
        ag[lane + 32 * c] = r;
    }
}

// ---------------------------------------------------------------------------
// Fused SAGE GEMM on V_WMMA_F32_16X16X4_F32 (true fp32 matrix op).
//   out[16 x 16*NT tile] = Aagg @ Wl + Aself @ Wr + bias   (optional ReLU)
// One wave per tile; wave-uniform bounds check keeps EXEC all-ones.
// A 16x4 f32 layout: lanes 0-15 hold K={0,1}, lanes 16-31 K={2,3}.
// B fragments come from K-pair-interleaved weights -> single b64 load.
// C/D: VGPR r is row M = r + 8*(lane>=16), col = lane%16.
// ---------------------------------------------------------------------------
template <int K, int NCOLS, int NT, bool RELU>
__global__ void __launch_bounds__(256)
sage_gemm_kernel(const float* __restrict__ Aagg,
                 const float* __restrict__ Aself,
                 const float* __restrict__ Pl,   // packed Wl [K/2][NCOLS][2]
                 const float* __restrict__ Pr,   // packed Wr
                 const float* __restrict__ bias,
                 float* __restrict__ out,
                 int total_tiles) {
    int wave = (blockIdx.x * blockDim.x + threadIdx.x) >> 5;
    if (wave >= total_tiles) return;
    int lane = threadIdx.x & 31;
    int half = lane >> 4;
    int lrow = lane & 15;

    constexpr int TILES_N = NCOLS / (16 * NT);
    int mt = wave / TILES_N;
    int nt = wave % TILES_N;
    int m0 = mt * 16;
    int n0 = nt * 16 * NT;
    int row = m0 + lrow;

    v8f acc[NT];
#pragma unroll
    for (int t = 0; t < NT; ++t) {
        float bv = bias[n0 + t * 16 + lrow];
        acc[t] = (v8f){bv, bv, bv, bv, bv, bv, bv, bv};
    }

    const float* a1 = Aagg + (size_t)row * K;
    const float* a2 = Aself + (size_t)row * K;

    // Pass 1: mean-aggregated neighbors through Wl
    for (int k0 = 0; k0 < K; k0 += 4) {
        int ka = k0 + 2 * half;                    // even
        v2f a = *(const v2f*)(a1 + ka);
#pragma unroll
        for (int t = 0; t < NT; ++t) {
            int col = n0 + t * 16 + lrow;
            v2f b = *(const v2f*)(Pl + ((size_t)(ka >> 1) * NCOLS + col) * 2);
            acc[t] = __builtin_amdgcn_wmma_f32_16x16x4_f32(
                false, a, false, b, (short)0, acc[t], false, false);
        }
    }
    // Pass 2: self features through Wr
    for (int k0 = 0; k0 < K; k0 += 4) {
        int ka = k0 + 2 * half;
        v2f a = *(const v2f*)(a2 + ka);
#pragma unroll
        for (int t = 0; t < NT; ++t) {
            int col = n0 + t * 16 + lrow;
            v2f b = *(const v2f*)(Pr + ((size_t)(ka >> 1) * NCOLS + col) * 2);
            acc[t] = __builtin_amdgcn_wmma_f32_16x16x4_f32(
                false, a, false, b, (short)0, acc[t], false, false);
        }
    }

#pragma unroll
    for (int t = 0; t < NT; ++t) {
#pragma unroll
        for (int r = 0; r < 8; ++r) {
            float v = acc[t][r];
            if (RELU) v = fmaxf(v, 0.0f);
            out[(size_t)(m0 + r + 8 * half) * NCOLS + n0 + t * 16 + lrow] = v;
        }
    }
}

// ---------------------------------------------------------------------------
// Launch
// ---------------------------------------------------------------------------
extern "C" void kernel_launch(void* const* d_in, const int* in_sizes, int n_in,
                              void* d_out, int out_size, void* d_ws,
                              size_t ws_size, hipStream_t stream) {
    const float*     x    = (const float*)d_in[0];
    const long long* edge = (const long long*)d_in[1];
    const float*     W1l  = (const float*)d_in[2];
    const float*     b1   = (const float*)d_in[3];
    const float*     W1r  = (const float*)d_in[4];
    const float*     W2l  = (const float*)d_in[5];
    const float*     b2   = (const float*)d_in[6];
    const float*     W2r  = (const float*)d_in[7];
    float*           out  = (float*)d_out;

    constexpr int DIN = 128, DH = 256, DOUT = 128;
    const int E = in_sizes[1] / 2;     // edge_index is [2, E]
    const int N = in_sizes[0] / DIN;   // 50000 = 16 * 3125 (exact)

    const long long* src = edge;
    const long long* dst = edge + E;

    // ---- workspace carving (256B-aligned sections) ----
    char* base = (char*)d_ws;
    size_t off = 0;
    auto carve = [&](size_t bytes) {
        char* p = base + off;
        off += (bytes + 255) & ~(size_t)255;
        return p;
    };
    int*   cnt      = (int*)carve((size_t)N * 4);
    int*   offs     = (int*)carve((size_t)N * 4);
    int*   cursor   = (int*)carve((size_t)N * 4);
    int*   blocksum = (int*)carve(128 * 4);
    int*   eidx     = (int*)carve((size_t)E * 4);
    float* agg1     = (float*)carve((size_t)N * DIN * 4);
    float* h        = (float*)carve((size_t)N * DH * 4);
    float* agg2     = (float*)carve((size_t)N * DH * 4);
    float* pW1l     = (float*)carve((size_t)DIN * DH * 4);
    float* pW1r     = (float*)carve((size_t)DIN * DH * 4);
    float* pW2l     = (float*)carve((size_t)DH * DOUT * 4);
    float* pW2r     = (float*)carve((size_t)DH * DOUT * 4);

    // ---- CSR build: histogram -> exclusive scan -> placement ----
    zero_int_kernel<<<(N + 255) / 256, 256, 0, stream>>>(cnt, N);
    zero_int_kernel<<<(N + 255) / 256, 256, 0, stream>>>(cursor, N);
    hist_kernel<<<(E + 255) / 256, 256, 0, stream>>>(dst, cnt, E);

    int nblk = (N + 1023) / 1024;  // 49 <= 128
    scan_block_kernel<<<nblk, 1024, 0, stream>>>(cnt, offs, blocksum, N);
    scan_top_kernel<<<1, 128, 0, stream>>>(blocksum, nblk);
    scan_add_kernel<<<nblk, 1024, 0, stream>>>(offs, blocksum, N);
    place_kernel<<<(E + 255) / 256, 256, 0, stream>>>(src, dst, offs, cursor,
                                                      eidx, E);

    // ---- weight repack (K-pair interleave for b64 B-fragment loads) ----
    repack_kernel<<<(DIN * DH + 255) / 256, 256, 0, stream>>>(W1l, pW1l,
                                                              DIN * DH, DH);
    repack_kernel<<<(DIN * DH + 255) / 256, 256, 0, stream>>>(W1r, pW1r,
                                                              DIN * DH, DH);
    repack_kernel<<<(DH * DOUT + 255) / 256, 256, 0, stream>>>(W2l, pW2l,
                                                               DH * DOUT, DOUT);
    repack_kernel<<<(DH * DOUT + 255) / 256, 256, 0, stream>>>(W2r, pW2r,
                                                               DH * DOUT, DOUT);

    // ---- layer 1: gather-mean, fused GEMM + bias + ReLU -> h ----
    aggregate_kernel<DIN><<<(N + 7) / 8, 256, 0, stream>>>(x, eidx, offs, cnt,
                                                           agg1, N);
    int tiles1 = (N / 16) * (DH / 64);
    sage_gemm_kernel<DIN, DH, 4, true>
        <<<(tiles1 + 7) / 8, 256, 0, stream>>>(agg1, x, pW1l, pW1r, b1, h,
                                               tiles1);

    // ---- layer 2: gather-mean, fused GEMM + bias -> out ----
    aggregate_kernel<DH><<<(N + 7) / 8, 256, 0, stream>>>(h, eidx, offs, cnt,
                                                          agg2, N);
    int tiles2 = (N / 16) * (DOUT / 64);
    sage_gemm_kernel<DH, DOUT, 4, false>
        <<<(tiles2 + 7) / 8, 256, 0, stream>>>(agg2, h, pW2l, pW2r, b2, out,
                                               tiles2);
}